// MarketRelation_75857712382388
// MI455X (gfx1250) — compile-verified
//
#include <hip/hip_runtime.h>
#include <hip/hip_bf16.h>
#include <math.h>

// Problem constants (fixed by the reference).
static constexpr int NN = 512;   // batch
static constexpr int TT = 8;     // time steps
static constexpr int DD = 64;    // feature dim
static constexpr float ALPHA = 0.5f;
static constexpr float BETA  = 0.1f;

typedef __attribute__((ext_vector_type(2))) float v2f;
typedef __attribute__((ext_vector_type(8))) float v8f;

// ---------------------------------------------------------------------------
// Phase 0: w[t] = softmax(a)[t] * (1 + ALPHA*exp(-BETA*(T-1-t)))  (8 values)
// ---------------------------------------------------------------------------
__global__ void mr_weights_kernel(const float* __restrict__ a,
                                  float* __restrict__ wt) {
    if (threadIdx.x == 0) {
        float m = a[0];
        for (int t = 1; t < TT; ++t) m = fmaxf(m, a[t]);
        float e[TT];
        float s = 0.f;
        for (int t = 0; t < TT; ++t) { e[t] = expf(a[t] - m); s += e[t]; }
        const float inv = 1.f / s;
        for (int t = 0; t < TT; ++t) {
            const float decay = 1.f + ALPHA * expf(-BETA * (float)(TT - 1 - t));
            wt[t] = e[t] * inv * decay;
        }
    }
}

// ---------------------------------------------------------------------------
// Phase 1: L[t][n][h] = x[n][t][:] . W1a[:,h]
//          Rb[t][n][h] = x[n][t][:] . W1b[:,h] + b1[h]
// Full-precision fp32 matmul on the matrix pipe: V_WMMA_F32_16X16X4_F32.
// One wave per 16x16 tile of the flat (4096 x 64) GEMM; A fragments are
// loaded once and reused for both the W1a and W1b accumulators.
//
// A 16x4 f32 layout (ISA 7.12.2): lanes 0-15 hold K=0,1 in VGPR0/1,
//                                 lanes 16-31 hold K=2,3.
// B  4x16 f32 layout: VGPR v, half-wave kl -> K = 2v + kl, N = lane%16.
// C/D 16x16 f32: VGPR r -> M = r + 8*(lane/16), N = lane%16.
// ---------------------------------------------------------------------------
__global__ __launch_bounds__(256) void mr_gemm_kernel(
    const float* __restrict__ x,   // (N,T,D) -> flat row r = n*T + t
    const float* __restrict__ W1,  // (2D, D)
    const float* __restrict__ b1,  // (D)
    float* __restrict__ L,         // (T,N,D)
    float* __restrict__ Rb)        // (T,N,D)
{
    const int lane = threadIdx.x & 31;
    const int wave = threadIdx.x >> 5;
    const int tile = blockIdx.x * 8 + wave;   // 1024 tiles total
    const int tm    = tile >> 2;              // 256 M-tiles (16 rows each)
    const int hbase = (tile & 3) * 16;        // 4 N-tiles (16 cols each)
    const int m  = lane & 15;
    const int kl = lane >> 4;                 // half-wave select (0/1)

    const float* xrow = x + (size_t)(tm * 16 + m) * DD;

    v8f accL = {0.f, 0.f, 0.f, 0.f, 0.f, 0.f, 0.f, 0.f};
    v8f accR = {0.f, 0.f, 0.f, 0.f, 0.f, 0.f, 0.f, 0.f};

#pragma unroll
    for (int kb = 0; kb < DD; kb += 4) {
        // A fragment: contiguous K pair per half-wave -> one b64 load.
        v2f av = *(const v2f*)(xrow + kb + 2 * kl);
        // B fragments for W1a and W1b.
        v2f bL, bR;
        bL.x = W1[(kb + kl    ) * DD + hbase + m];
        bL.y = W1[(kb + kl + 2) * DD + hbase + m];
        bR.x = W1[(DD + kb + kl    ) * DD + hbase + m];
        bR.y = W1[(DD + kb + kl + 2) * DD + hbase + m];
        accL = __builtin_amdgcn_wmma_f32_16x16x4_f32(
            false, av, false, bL, (short)0, accL, false, false);
        accR = __builtin_amdgcn_wmma_f32_16x16x4_f32(
            false, av, false, bR, (short)0, accR, false, false);
    }

    const float b1v = b1[hbase + m];
#pragma unroll
    for (int vr = 0; vr < 8; ++vr) {
        const int row = tm * 16 + vr + kl * 8;  // flat row = n*T + t
        const int t = row & (TT - 1);
        const int n = row >> 3;                 // /TT
        const size_t off = (size_t)(t * NN + n) * DD + hbase + m;
        L[off]  = accL[vr];
        Rb[off] = accR[vr] + b1v;
    }
}

// ---------------------------------------------------------------------------
// Phase 2 (dominant): A[i][j] = sum_t w[t]*relu( relu(L_t[i]+Rb_t[j]).w2 + b2 )
// Nonlinearity between add and dot -> pure VALU workload.
// Lane owns column j: Rb row + w2 live in VGPRs. Wave owns 4 rows i:
// L rows are wave-uniform -> SGPR broadcast, so the inner loop is
// v_add(sgpr,vgpr) ; v_max(0) ; v_fmac -> 3 VALU per (t,i,j,d).
// Block = 8 waves sharing one 32-wide j tile (R rows hit in cache/L2).
// ---------------------------------------------------------------------------
__global__ __launch_bounds__(256) void mr_pair_kernel(
    const float* __restrict__ L,    // (T,N,D)
    const float* __restrict__ Rb,   // (T,N,D)  (already + b1)
    const float* __restrict__ W2,   // (D)
    const float* __restrict__ wt,   // (T)
    const float* __restrict__ b2p,  // (1)
    float* __restrict__ out)        // (N,N)
{
    const int lane = threadIdx.x & 31;
    const int wave = threadIdx.x >> 5;
    const int j  = blockIdx.x * 32 + lane;
    const int i0 = blockIdx.y * 32 + wave * 4;
    const float b2v = b2p[0];

    float w2r[DD];
#pragma unroll
    for (int d = 0; d < DD; d += 4) {
        const float4 v = *(const float4*)(W2 + d);
        w2r[d] = v.x; w2r[d + 1] = v.y; w2r[d + 2] = v.z; w2r[d + 3] = v.w;
    }

    float acc[4] = {0.f, 0.f, 0.f, 0.f};
    for (int t = 0; t < TT; ++t) {
        float r[DD];
        const float* Rrow = Rb + (size_t)(t * NN + j) * DD;
#pragma unroll
        for (int d = 0; d < DD; d += 4) {
            const float4 v = *(const float4*)(Rrow + d);
            r[d] = v.x; r[d + 1] = v.y; r[d + 2] = v.z; r[d + 3] = v.w;
        }
        const float wtv = wt[t];
#pragma unroll
        for (int ii = 0; ii < 4; ++ii) {
            const float* Lrow = L + (size_t)(t * NN + i0 + ii) * DD; // uniform
            float dot = b2v;
#pragma unroll
            for (int d = 0; d < DD; ++d)
                dot = __builtin_fmaf(__builtin_fmaxf(Lrow[d] + r[d], 0.f),
                                     w2r[d], dot);
            acc[ii] = __builtin_fmaf(wtv, __builtin_fmaxf(dot, 0.f), acc[ii]);
        }
    }
#pragma unroll
    for (int ii = 0; ii < 4; ++ii)
        out[(size_t)(i0 + ii) * NN + j] = acc[ii];
}

// ---------------------------------------------------------------------------
// Launch: phase0 (weights) -> phase1 (WMMA GEMM) -> phase2 (pair loop).
// Workspace: L (1 MB) + Rb (1 MB) + w (32 B)  << typical ws_size.
// All writes fully overwrite their ranges each call -> deterministic.
// ---------------------------------------------------------------------------
extern "C" void kernel_launch(void* const* d_in, const int* in_sizes, int n_in,
                              void* d_out, int out_size, void* d_ws,
                              size_t ws_size, hipStream_t stream) {
    const float* x  = (const float*)d_in[0];  // (512,8,64)
    const float* W1 = (const float*)d_in[1];  // (128,64)
    const float* b1 = (const float*)d_in[2];  // (64)
    const float* W2 = (const float*)d_in[3];  // (64,1)
    const float* b2 = (const float*)d_in[4];  // (1)
    const float* a  = (const float*)d_in[5];  // (8,1)
    float* out = (float*)d_out;               // (512,512)

    float* Lw = (float*)d_ws;                 // T*N*D floats
    float* Rw = Lw + (size_t)TT * NN * DD;    // T*N*D floats
    float* ww = Rw + (size_t)TT * NN * DD;    // T floats

    mr_weights_kernel<<<1, 32, 0, stream>>>(a, ww);
    mr_gemm_kernel<<<128, 256, 0, stream>>>(x, W1, b1, Lw, Rw);
    mr_pair_kernel<<<dim3(16, 16), 256, 0, stream>>>(Lw, Rw, W2, ww, b2, out);
}